// MeshConv_transpose_76819785056400
// MI455X (gfx1250) — compile-verified
//
#include <hip/hip_runtime.h>
#include <hip/hip_bf16.h>

#define NV      40962
#define NVP     10242
#define NB      8
#define TILES   ((NV + 15) / 16)        // 2561 vertex tiles per batch
#define TOT     (NB * TILES)            // 20488 wave-tiles
#define WAVES   4                       // waves per block
#define ITERS   ((TOT + WAVES - 1) / WAVES)   // 5122 block iterations
#define WS      260                     // LDS row stride (floats) for Wt, padded
#define FS      260                     // LDS row stride (floats) for F, padded

typedef __attribute__((ext_vector_type(2))) float v2f;
typedef __attribute__((ext_vector_type(8))) float v8f;

// ---------------- Kernel 1: transpose x[b][c][n] -> xt[b][n][c] ----------------
__global__ void __launch_bounds__(256)
transpose_kernel(const float* __restrict__ x, float* __restrict__ xt) {
    __shared__ float tile[64][65];
    int bid = blockIdx.x;
    int b   = bid / 161;            // ceil(10242/64) = 161 tiles per batch
    int n0  = (bid % 161) * 64;
    int tid = threadIdx.x;
    for (int i = tid; i < 64 * 64; i += 256) {
        int c = i >> 6, nn = i & 63;
        int n = n0 + nn;
        tile[c][nn] = (n < NVP) ? x[((size_t)b * 64 + c) * NVP + n] : 0.0f;
    }
    __syncthreads();
    for (int i = tid; i < 64 * 64; i += 256) {
        int nn = i >> 6, c = i & 63;
        int n = n0 + nn;
        if (n < NVP) xt[((size_t)b * NVP + n) * 64 + c] = tile[c][nn];
    }
}

// ---------------- sparse branch: F[r][dst + c] = sum_k val * xt[col][c] --------
__device__ __forceinline__ void build_branch(float* F, const float* __restrict__ xt,
                                             const int* __restrict__ cols,
                                             const float* __restrict__ vals,
                                             int K, int dst, int b, int n0, int lane) {
    for (int r = 0; r < 16; ++r) {
        int n = n0 + r;
        v2f acc; acc.x = 0.0f; acc.y = 0.0f;
        if (n < NV) {
            if (n + 1 < NV) __builtin_prefetch(&cols[(size_t)(n + 1) * K], 0, 0);
            for (int k = 0; k < K; ++k) {
                int   col = cols[(size_t)n * K + k];
                float v   = vals[(size_t)n * K + k];
                if (col < NVP) {
                    v2f g = *(const v2f*)&xt[((size_t)b * NVP + col) * 64 + 2 * lane];
                    acc.x += v * g.x;
                    acc.y += v * g.y;
                } else {               // padded vertices are ones
                    acc.x += v;
                    acc.y += v;
                }
            }
        }
        *(v2f*)&F[r * FS + dst + 2 * lane] = acc;   // contiguous b64, conflict-free
    }
}

// ---------------- Kernel 2: gather + fused [M,256]x[256,64] WMMA GEMM ----------
__global__ void __launch_bounds__(128)
meshconv_kernel(const int* __restrict__ Lc, const float* __restrict__ Lv,
                const int* __restrict__ Ec, const float* __restrict__ Ev,
                const int* __restrict__ Nc, const float* __restrict__ Nvv,
                const float* __restrict__ coeffs, const float* __restrict__ bias,
                const float* __restrict__ xt, float* __restrict__ out) {
    __shared__ float Wt[64 * WS];          // Wt[c_out][k], k = branch*64 + c_in
    __shared__ float Fb[WAVES][16 * FS];   // per-wave 16-row feature tile
    __shared__ float bsh[64];

    int tid = threadIdx.x;
    // stacked coefficient matrix, transposed: Wt[c][b*64+ci] = coeffs[b][ci][c]
    for (int i = tid; i < 64 * 256; i += 128) {
        int c = i & 63, k = i >> 6;                 // consecutive tid -> coalesced read
        Wt[c * WS + k] = coeffs[(size_t)k * 64 + c];
    }
    if (tid < 64) bsh[tid] = bias[tid];
    __syncthreads();

    int wave = tid >> 5;
    int lane = tid & 31;
    int half = lane >> 4;        // 0: K=k0,k0+1   1: K=k0+2,k0+3 (ISA 32-bit A/B layout)
    int r16  = lane & 15;

    v8f biasv[4];                // D-tile layout: vgpr j -> c = m*16 + j + 8*half
    for (int m = 0; m < 4; ++m)
        for (int j = 0; j < 8; ++j)
            biasv[m][j] = bsh[m * 16 + j + 8 * half];

    float* F = Fb[wave];

    for (int it = blockIdx.x; it < ITERS; it += gridDim.x) {
        int  t      = it * WAVES + wave;
        bool tvalid = t < TOT;
        int  tc     = tvalid ? t : TOT - 1;         // clamp so barriers stay uniform
        int  b      = tc / TILES;
        int  n0     = (tc % TILES) * 16;

        // ----- build F[16][256]: identity | laplacian | grad_ew | grad_ns -----
        for (int r = 0; r < 16; ++r) {              // identity from xt (coalesced)
            int n = n0 + r;
            v2f v;
            if (n < NVP) v = *(const v2f*)&xt[((size_t)b * NVP + n) * 64 + 2 * lane];
            else         { v.x = 1.0f; v.y = 1.0f; }
            *(v2f*)&F[r * FS + 2 * lane] = v;
        }
        build_branch(F, xt, Lc, Lv, 7,  64,  b, n0, lane);
        build_branch(F, xt, Ec, Ev, 18, 128, b, n0, lane);
        build_branch(F, xt, Nc, Nvv, 18, 192, b, n0, lane);

        __syncthreads();

        // ----- D(64x16) = Wt(64x256) x F^T(256x16) + bias, fp32 WMMA ----------
        v8f acc0 = biasv[0], acc1 = biasv[1], acc2 = biasv[2], acc3 = biasv[3];
        for (int k0 = 0; k0 < 256; k0 += 4) {
            int ko = k0 + 2 * half;
            v2f bf = *(const v2f*)&F[r16 * FS + ko];                 // B: F^T frag
            v2f a0 = *(const v2f*)&Wt[(0 * 16 + r16) * WS + ko];     // A: Wt frags
            v2f a1 = *(const v2f*)&Wt[(1 * 16 + r16) * WS + ko];
            v2f a2 = *(const v2f*)&Wt[(2 * 16 + r16) * WS + ko];
            v2f a3 = *(const v2f*)&Wt[(3 * 16 + r16) * WS + ko];
            acc0 = __builtin_amdgcn_wmma_f32_16x16x4_f32(false, a0, false, bf, (short)0, acc0, false, false);
            acc1 = __builtin_amdgcn_wmma_f32_16x16x4_f32(false, a1, false, bf, (short)0, acc1, false, false);
            acc2 = __builtin_amdgcn_wmma_f32_16x16x4_f32(false, a2, false, bf, (short)0, acc2, false, false);
            acc3 = __builtin_amdgcn_wmma_f32_16x16x4_f32(false, a3, false, bf, (short)0, acc3, false, false);
        }

        // ----- store out[b][c][n]: lanes span 16 consecutive n -> coalesced ---
        int n = n0 + r16;
        if (tvalid && n < NV) {
            for (int j = 0; j < 8; ++j) {
                int c0 = j + 8 * half;
                out[((size_t)b * 64 + c0 +  0) * NV + n] = acc0[j];
                out[((size_t)b * 64 + c0 + 16) * NV + n] = acc1[j];
                out[((size_t)b * 64 + c0 + 32) * NV + n] = acc2[j];
                out[((size_t)b * 64 + c0 + 48) * NV + n] = acc3[j];
            }
        }
        __syncthreads();   // protect F against next iteration's writes
    }
}

extern "C" void kernel_launch(void* const* d_in, const int* in_sizes, int n_in,
                              void* d_out, int out_size, void* d_ws, size_t ws_size,
                              hipStream_t stream) {
    const float* x      = (const float*)d_in[0];
    const int*   Lc     = (const int*)  d_in[1];
    const float* Lv     = (const float*)d_in[2];
    const int*   Ec     = (const int*)  d_in[3];
    const float* Ev     = (const float*)d_in[4];
    const int*   Nc     = (const int*)  d_in[5];
    const float* Nvv    = (const float*)d_in[6];
    const float* coeffs = (const float*)d_in[7];
    const float* bias   = (const float*)d_in[8];
    float*       out    = (float*)d_out;
    float*       xt     = (float*)d_ws;   // needs 8*10242*64*4 = ~21 MB

    transpose_kernel<<<NB * 161, 256, 0, stream>>>(x, xt);
    meshconv_kernel<<<1280, 128, 0, stream>>>(Lc, Lv, Ec, Ev, Nc, Nvv,
                                              coeffs, bias, xt, out);
}